// GATLayer_43164421325633
// MI455X (gfx1250) — compile-verified
//
#include <hip/hip_runtime.h>
#include <math.h>

// Problem constants (fixed by the reference): B=4, N=512, d=128, H=8, Hd=16.
#define BB 4
#define NN 512
#define DD 128
#define HH 8
#define HD 16
#define BN (BB * NN)          // 2048 rows total
#define E_STRIDE 132          // LDS row stride (floats) to kill bank conflicts
#define TROWS 32              // e-rows per TDM tile
#define NPASS (NN / TROWS)    // 16 passes

typedef float v2f __attribute__((ext_vector_type(2)));
typedef float v4f __attribute__((ext_vector_type(4)));
typedef float v8f __attribute__((ext_vector_type(8)));
typedef unsigned int u32x4 __attribute__((ext_vector_type(4)));
typedef int i32x4 __attribute__((ext_vector_type(4)));
typedef int i32x8 __attribute__((ext_vector_type(8)));

__device__ __forceinline__ v8f wmma_f32(v2f a, v2f b, v8f c) {
  // V_WMMA_F32_16X16X4_F32 : D = A(16x4) * B(4x16) + C(16x16), all f32
  return __builtin_amdgcn_wmma_f32_16x16x4_f32(false, a, false, b, (short)0, c,
                                               false, false);
}

// ---------------------------------------------------------------------------
// TDM: issue a 2D tensor load of a TROWSx128 f32 tile into LDS.
// D# pad feature inserts 4 DWORDs after every 128 DWORDs -> LDS row stride 132.
// Executed once per wave (gate at wave granularity!).
// ---------------------------------------------------------------------------
__device__ __forceinline__ void tdm_load_e_tile(const float* gsrc,
                                                const void* ldst) {
  unsigned long long ga = (unsigned long long)(uintptr_t)gsrc;
  unsigned lds = (unsigned)(uintptr_t)ldst;  // LDS aperture: low 32 bits = offset
  u32x4 g0;
  g0[0] = 1u;                                     // count=1 (valid descriptor)
  g0[1] = lds;                                    // lds_addr (bytes)
  g0[2] = (unsigned)(ga & 0xFFFFFFFFull);         // global_addr[31:0]
  g0[3] = (unsigned)((ga >> 32) & 0x1FFFFFFull)   // global_addr[56:32]
          | (2u << 30);                           // type=2 ("image")
  i32x8 g1;
  g1[0] = (int)((2u << 16)    // data_size = 4 bytes
                | (1u << 20)  // pad_enable
                | (6u << 22)  // pad_interval code 6 -> every 128 DWORDs
                | (3u << 25)); // pad_amount  code 3 -> 4 DWORDs
  g1[1] = (int)(128u << 16);                  // tensor_dim0 = 128 (lo16)
  g1[2] = (int)((unsigned)TROWS << 16);       // dim0 hi16=0 | tensor_dim1 lo16
  g1[3] = (int)(128u << 16);                  // dim1 hi16=0 | tile_dim0 = 128
  g1[4] = (int)TROWS;                         // tile_dim1 = TROWS, tile_dim2=0
  g1[5] = 128;                                // tensor_dim0_stride lo32
  g1[6] = 0;
  g1[7] = 0;
  i32x4 z4 = {0, 0, 0, 0};
#if __clang_major__ >= 23
  i32x8 z8 = {0, 0, 0, 0, 0, 0, 0, 0};
  __builtin_amdgcn_tensor_load_to_lds(g0, g1, z4, z4, z8, 0);
#else
  __builtin_amdgcn_tensor_load_to_lds(g0, g1, z4, z4, 0);
#endif
}

// ---------------------------------------------------------------------------
// K1: Q/K/V projections. (2048x128) @ (128x128) per weight, f32 WMMA.
// grid = (BN/16, 3), block = 256 (8 waves); wave w owns column tile w*16.
// ---------------------------------------------------------------------------
__global__ __launch_bounds__(256) void qkv_kernel(
    const float* __restrict__ h, const float* __restrict__ Wq,
    const float* __restrict__ Wk, const float* __restrict__ Wv,
    float* __restrict__ ws) {
  const int w = threadIdx.x >> 5, lane = threadIdx.x & 31;
  const int lo = lane & 15, hi = lane >> 4;
  const int rt = blockIdx.x * 16;
  const int ct = w * 16;
  const float* W;
  float* out;
  if (blockIdx.y == 0)      { W = Wq; out = ws; }
  else if (blockIdx.y == 1) { W = Wk; out = ws + BN * DD; }
  else                      { W = Wv; out = ws + 2 * BN * DD; }

  const int row = rt + lo;
  v8f c = {0.f, 0.f, 0.f, 0.f, 0.f, 0.f, 0.f, 0.f};
#pragma unroll
  for (int k = 0; k < DD; k += 4) {
    const int ka = k + 2 * hi;                       // K index for this lane
    v2f a = *(const v2f*)(h + row * DD + ka);        // A[M=lo, K=ka..ka+1]
    v2f b;
    b.x = W[ka * DD + ct + lo];                      // B[K=ka,   N=lo]
    b.y = W[(ka + 1) * DD + ct + lo];                // B[K=ka+1, N=lo]
    c = wmma_f32(a, b, c);
  }
#pragma unroll
  for (int v = 0; v < 8; ++v)                        // D: M = v + 8*hi, N = lo
    out[(rt + v + 8 * hi) * DD + ct + lo] = c[v];
}

// ---------------------------------------------------------------------------
// K2: per-(b,n) attention row. e[b,n,:,:] is staged by the Tensor Data Mover
// into double-buffered LDS (DMA of tile p+1 overlaps WMMA of tile p), then:
//   LDS -> out_e passthrough store, WMMA edge-bias, QK^T, softmax, attn*V,
//   Wo projection + ReLU -> h_proj.  grid = BN blocks, block = 256 (8 waves).
// ---------------------------------------------------------------------------
__global__ __launch_bounds__(256) void attn_kernel(
    const float* __restrict__ e, const float* __restrict__ We,
    const float* __restrict__ Wo, const float* __restrict__ bo,
    const float* __restrict__ ws, float* __restrict__ hp,
    float* __restrict__ out_e) {
  __shared__ float e_buf[2][TROWS * E_STRIDE];  // 33792 B (reused as scratch)
  __shared__ float scores[HH * NN];             // 16384 B
  __shared__ float part[8 * 256];               //  8192 B (per-wave D tiles)
  __shared__ float q_row[DD];                   //   512 B
  __shared__ float out_row[DD];                 //   512 B

  const int t = threadIdx.x;
  const int w = t >> 5, lane = t & 31;
  const int lo = lane & 15, hi = lane >> 4;
  const int bn = blockIdx.x;               // b*N + n
  const int b = bn >> 9;                   // /512

  const float* Q = ws;
  const float* Kp = ws + BN * DD;
  const float* Vp = ws + 2 * BN * DD;

  if (t < 32) ((v4f*)q_row)[t] = ((const v4f*)(Q + (size_t)bn * DD))[t];

  const size_t ebase = (size_t)bn * (NN * DD);

  // ---- edge bias via WMMA; TDM double-buffered e streaming ----
  const int mtile = (w & 1) * 16;   // m sub-tile inside pass (2 tiles)
  const int koff = (w >> 1) * 32;   // K quarter (split across 4 wave pairs)

  if (w == 0) tdm_load_e_tile(e + ebase, &e_buf[0][0]);  // prologue: pass 0
  int cur = 0;
  for (int pass = 0; pass < NPASS; ++pass) {
    if (w == 0) {
      if (pass + 1 < NPASS) {
        tdm_load_e_tile(e + ebase + (size_t)(pass + 1) * TROWS * DD,
                        &e_buf[cur ^ 1][0]);
        __builtin_amdgcn_s_wait_tensorcnt(1);  // oldest (current tile) done
      } else {
        __builtin_amdgcn_s_wait_tensorcnt(0);
      }
    }
    __syncthreads();  // publish TDM-written tile to all waves
    const float* et = &e_buf[cur][0];

    // passthrough: LDS -> out_e (e touched once in HBM each direction)
#pragma unroll
    for (int i = 0; i < 4; ++i) {
      int f4 = t + 256 * i;              // 0..1023 float4 chunks
      int r = f4 >> 5, c4 = f4 & 31;
      v4f val = *(const v4f*)(et + r * E_STRIDE + c4 * 4);
      *(v4f*)(out_e + ebase + (size_t)(pass * TROWS + r) * DD + c4 * 4) = val;
    }

    // bias_part(16x16) = e_tile[mtile.., koff..koff+32) @ We (padded to 16)
    v8f c = {0.f, 0.f, 0.f, 0.f, 0.f, 0.f, 0.f, 0.f};
#pragma unroll
    for (int k2 = 0; k2 < 8; ++k2) {
      int k = koff + k2 * 4 + 2 * hi;
      v2f a = *(const v2f*)(et + (mtile + lo) * E_STRIDE + k);
      v2f bb;
      bb.x = (lo < HH) ? We[k * HH + lo] : 0.f;
      bb.y = (lo < HH) ? We[(k + 1) * HH + lo] : 0.f;
      c = wmma_f32(a, bb, c);
    }
#pragma unroll
    for (int v = 0; v < 8; ++v) part[w * 256 + v * 32 + lane] = c[v];
    __syncthreads();

    // combine the 4 K-quarters -> scores[h][m]
    {
      int mrow = t & 31, hh = t >> 5;   // 32 rows x 8 heads = 256 threads
      int tile = mrow >> 4, r = mrow & 15;
      int l = hh + ((r >= 8) ? 16 : 0); // D layout: M = v + 8*(l>>4), N = l&15
      int v = r & 7;
      float s = 0.f;
#pragma unroll
      for (int q = 0; q < 4; ++q) s += part[(tile + 2 * q) * 256 + v * 32 + l];
      scores[hh * NN + pass * TROWS + mrow] = s;
    }
    __syncthreads();  // part + e_buf[cur] free for reuse
    cur ^= 1;
  }

  // ---- scores += (1/sqrt(Hd)) * Q·K, per (h,m) ----
#pragma unroll
  for (int j = 0; j < 16; ++j) {
    int idx = t + 256 * j;               // 0..4095 = 512 m x 8 heads
    int hh = idx & 7, m = idx >> 3;
    const float* kr = Kp + (size_t)(b * NN + m) * DD + hh * HD;
    const float* qr = q_row + hh * HD;
    float acc = 0.f;
#pragma unroll
    for (int u = 0; u < 4; ++u) {
      v4f kv = *(const v4f*)(kr + 4 * u);
      acc += qr[4 * u + 0] * kv.x + qr[4 * u + 1] * kv.y +
             qr[4 * u + 2] * kv.z + qr[4 * u + 3] * kv.w;
    }
    scores[hh * NN + m] += 0.25f * acc;
  }
  __syncthreads();

  // ---- softmax: wave w owns head w; lane covers m = lane + 32*i (wave32) ----
  float vals[16];
  float mx = -3.4e38f;
#pragma unroll
  for (int i = 0; i < 16; ++i) {
    vals[i] = scores[w * NN + lane + 32 * i];
    mx = fmaxf(mx, vals[i]);
  }
#pragma unroll
  for (int s = 16; s >= 1; s >>= 1) mx = fmaxf(mx, __shfl_xor(mx, s, 32));
  float sum = 0.f;
#pragma unroll
  for (int i = 0; i < 16; ++i) {
    vals[i] = __expf(vals[i] - mx);
    sum += vals[i];
  }
#pragma unroll
  for (int s = 16; s >= 1; s >>= 1) sum += __shfl_xor(sum, s, 32);
  const float inv = 1.f / sum;

  // ---- out[h, 0:16] = sum_m attn[h,m] * V[b,m,h,:] ----
  float acc[16];
#pragma unroll
  for (int k = 0; k < 16; ++k) acc[k] = 0.f;
  const float* Vb = Vp + (size_t)(b * NN) * DD + w * HD;
  for (int i = 0; i < 16; ++i) {
    int m = lane + 32 * i;
    float p = vals[i] * inv;
    const float* vr = Vb + (size_t)m * DD;
#pragma unroll
    for (int u = 0; u < 4; ++u) {
      v4f vv = *(const v4f*)(vr + 4 * u);
      acc[4 * u + 0] += p * vv.x;
      acc[4 * u + 1] += p * vv.y;
      acc[4 * u + 2] += p * vv.z;
      acc[4 * u + 3] += p * vv.w;
    }
  }
  float* red = &e_buf[0][0];  // reuse (e staging done)
#pragma unroll
  for (int k = 0; k < 16; ++k) red[(w * 32 + lane) * 16 + k] = acc[k];
  __syncthreads();
  if (t < 128) {
    int hh = t >> 4, k = t & 15;
    float s = 0.f;
#pragma unroll
    for (int l = 0; l < 32; ++l) s += red[(hh * 32 + l) * 16 + k];
    out_row[hh * HD + k] = s;
  }
  __syncthreads();

  // ---- projection: relu(out_row @ Wo + bo) -> h_proj row ----
  if (t < 128) {
    float a2 = bo[t];
#pragma unroll 8
    for (int dd2 = 0; dd2 < 128; ++dd2) a2 += out_row[dd2] * Wo[dd2 * DD + t];
    hp[(size_t)bn * DD + t] = fmaxf(a2, 0.f);
  }
}

// ---------------------------------------------------------------------------
// K3a: per-block partial sums/sumsq over 32 rows (deterministic, no atomics).
// grid = 64, block = 256. partial[blk][0:128]=sum, [128:256]=sumsq.
// ---------------------------------------------------------------------------
__global__ __launch_bounds__(256) void bn_stats(const float* __restrict__ hp,
                                                float* __restrict__ partial) {
  __shared__ float ls[256], lq[256];
  int t = threadIdx.x, f = t & 127, half = t >> 7;
  float s = 0.f, q = 0.f;
  int base = blockIdx.x * 32;
#pragma unroll
  for (int j = 0; j < 16; ++j) {
    int row = base + half + 2 * j;
    float v = hp[(size_t)row * DD + f];
    s += v;
    q += v * v;
  }
  ls[t] = s;
  lq[t] = q;
  __syncthreads();
  if (t < 128) {
    partial[blockIdx.x * 256 + t] = ls[t] + ls[t + 128];
    partial[blockIdx.x * 256 + 128 + t] = lq[t] + lq[t + 128];
  }
}

// K3b: reduce 64 partials -> stats[0:128]=sum, stats[128:256]=sumsq.
__global__ void bn_reduce(const float* __restrict__ partial,
                          float* __restrict__ stats) {
  int t = threadIdx.x;  // 256 threads
  float s = 0.f;
#pragma unroll 8
  for (int blk = 0; blk < 64; ++blk) s += partial[blk * 256 + t];
  stats[t] = s;
}

// ---------------------------------------------------------------------------
// K4: BN (batch stats, eps=1e-5) + residual -> h_new. grid = BN*DD/256.
// ---------------------------------------------------------------------------
__global__ __launch_bounds__(256) void bn_finish(
    const float* __restrict__ hp, const float* __restrict__ stats,
    const float* __restrict__ gamma, const float* __restrict__ beta,
    const float* __restrict__ hin, float* __restrict__ out) {
  int idx = blockIdx.x * 256 + threadIdx.x;
  int f = idx & 127;
  float mean = stats[f] * (1.f / (float)BN);
  float var = stats[128 + f] * (1.f / (float)BN) - mean * mean;
  float x = hp[idx];
  out[idx] =
      gamma[f] * (x - mean) * rsqrtf(var + 1e-5f) + beta[f] + hin[idx];
}

// ---------------------------------------------------------------------------
extern "C" void kernel_launch(void* const* d_in, const int* in_sizes, int n_in,
                              void* d_out, int out_size, void* d_ws,
                              size_t ws_size, hipStream_t stream) {
  (void)in_sizes; (void)n_in; (void)out_size; (void)ws_size;
  const float* h     = (const float*)d_in[0];
  const float* e     = (const float*)d_in[1];
  const float* Wq    = (const float*)d_in[2];
  const float* Wk    = (const float*)d_in[3];
  const float* Wv    = (const float*)d_in[4];
  const float* We    = (const float*)d_in[5];
  const float* Wo    = (const float*)d_in[6];
  const float* bo    = (const float*)d_in[7];
  const float* gamma = (const float*)d_in[8];
  const float* beta  = (const float*)d_in[9];

  float* out   = (float*)d_out;
  float* out_h = out;                 // h_new: BN*DD floats
  float* out_e = out + BN * DD;       // e passthrough: B*N*N*d floats

  // Workspace layout (floats): Q | K | V | h_proj | partials(64*256) | stats(256)
  float* ws      = (float*)d_ws;
  float* hp      = ws + 3 * BN * DD;
  float* partial = ws + 4 * BN * DD;
  float* stats   = partial + 64 * 256;

  qkv_kernel<<<dim3(BN / 16, 3), 256, 0, stream>>>(h, Wq, Wk, Wv, ws);
  attn_kernel<<<BN, 256, 0, stream>>>(e, We, Wo, bo, ws, hp, out_e);
  bn_stats<<<BN / 32, 256, 0, stream>>>(hp, partial);
  bn_reduce<<<1, 256, 0, stream>>>(partial, stats);
  bn_finish<<<(BN * DD) / 256, 256, 0, stream>>>(hp, stats, gamma, beta, h,
                                                 out_h);
}